// JpegLayer_65996467470858
// MI455X (gfx1250) — compile-verified
//
#include <hip/hip_runtime.h>
#include <hip/hip_bf16.h>

// ===========================================================================
// JPEG layer (RGB->YCbCr, 2x2 chroma pool, 8x8 DCT, quantize+round) for
// MI455X / gfx1250.  Memory-bound: ~302 MB traffic -> ~13us @ 23.3 TB/s.
// DCT runs on the WMMA pipe: Dblk = diag(D8,D8) (16x16), so one pair of
// 16x16 WMMA passes performs FOUR independent 8x8 DCTs:
//   S   = Dblk * Tile      (pass 1, A = Dblk fragment, B = tile from LDS)
//   R^T = Dblk * S^T       (pass 2, S transposed through LDS)
// Uses V_WMMA_F32_16X16X4_F32 (fp32-exact, K accumulated in 4 chunks).
// Quantizer reciprocals are precomputed once per block into LDS so the hot
// path is v_mul + v_rndne instead of IEEE-division microcode.
// ===========================================================================

typedef __attribute__((ext_vector_type(2)))  float    v2f;
typedef __attribute__((ext_vector_type(4)))  float    f4;
typedef __attribute__((ext_vector_type(8)))  float    v8f;

// Only check on the device pass: on the host pass the builtin is an
// aux-target builtin (parseable in __device__ code but invisible to
// __has_builtin against the x86 target).
#if defined(__HIP_DEVICE_COMPILE__)
#  if !__has_builtin(__builtin_amdgcn_wmma_f32_16x16x4_f32)
#    error "expected __builtin_amdgcn_wmma_f32_16x16x4_f32 on gfx1250"
#  endif
#endif

#define JPEG_OFFSET (128.0f / 255.0f)

// 8x8 DCT-II matrix element (orthonormal), row i col j
__device__ __forceinline__ float dct8(int i, int j) {
    if (i == 0) return 0.35355339059327379f;             // 1/sqrt(8)
    return 0.5f * __cosf(0.19634954084936207f * (float)((2 * j + 1) * i)); // pi/16
}
// Block-diagonal 16x16: diag(D8, D8)
__device__ __forceinline__ float dblk(int m, int c) {
    if ((m >> 3) != (c >> 3)) return 0.0f;
    return dct8(m & 7, c & 7);
}

struct AFrag { v2f a[4]; };   // A 16x4 fragment per K-chunk: 2 VGPRs each

__device__ __forceinline__ AFrag make_afrag(int lane) {
    AFrag f;
    const int m = lane & 15, hi = lane >> 4;
    // 32-bit A 16x4 layout: lanes0-15 hold K = {4kk, 4kk+1}, lanes16-31 K = {4kk+2, 4kk+3}
#pragma unroll
    for (int kk = 0; kk < 4; ++kk) {
        const int c0 = 4 * kk + 2 * hi;
        f.a[kk].x = dblk(m, c0);
        f.a[kk].y = dblk(m, c0 + 1);
    }
    return f;
}

// acc = Dblk (16x16) * src (16x16, LDS, row stride `stride` floats)
__device__ __forceinline__ v8f mm_dblk(const AFrag& A, const float* src, int stride, int lane) {
    const int c = lane & 15, hi = lane >> 4;
    v8f acc = {};
#pragma unroll
    for (int kk = 0; kk < 4; ++kk) {
        const int k0 = 4 * kk + 2 * hi;   // B 4x16: lanes0-15 rows {k,k+1}, lanes16-31 {k+2,k+3}
        v2f b;
        b.x = src[k0 * stride + c];
        b.y = src[(k0 + 1) * stride + c];
        acc = __builtin_amdgcn_wmma_f32_16x16x4_f32(false, A.a[kk], false, b,
                                                    (short)0, acc, false, false);
    }
    return acc;
}

// Full 16x16 two-sided DCT: returns R^T fragment.
// Lane L holds R[L&15][8*(L>>4)+v] for v = 0..7 (one 8-wide block row).
__device__ __forceinline__ v8f tile_dct(const AFrag& A, const float* src, int stride,
                                        float* St, int lane) {
    const int c = lane & 15, hi = lane >> 4;
    v8f s = mm_dblk(A, src, stride, lane);             // S = Dblk * X
    // Lane holds S[v + 8*hi][c]; store transposed: St[c][v + 8*hi]
    float* d = St + c * 16 + 8 * hi;
    *(f4*)(d)     = f4{s[0], s[1], s[2], s[3]};
    *(f4*)(d + 4) = f4{s[4], s[5], s[6], s[7]};
    __builtin_amdgcn_wave_barrier();                   // DS ops in-order per wave
    v8f r = mm_dblk(A, St, 16, lane);                  // R^T = Dblk * S^T
    __builtin_amdgcn_wave_barrier();
    return r;
}

// out[j] = rintf(r[j] * qinv[j]);  rint == round-half-even == jnp.round
__device__ __forceinline__ void quant_store(v8f r, f4 q0, f4 q1, float* o) {
    f4 lo = { rintf(r[0] * q0.x), rintf(r[1] * q0.y),
              rintf(r[2] * q0.z), rintf(r[3] * q0.w) };
    f4 hi = { rintf(r[4] * q1.x), rintf(r[5] * q1.y),
              rintf(r[6] * q1.z), rintf(r[7] * q1.w) };
    *(f4*)(o)     = lo;
    *(f4*)(o + 4) = hi;
}

// One wave32 <-> one 32-wide x 16-tall source-pixel region.
//   -> 2 Y tiles (16x16 each = 4 Y blocks each)
//   -> 1 chroma tile (rows 0-7 = pooled Cb 8x16, rows 8-15 = pooled Cr 8x16)
__global__ __launch_bounds__(256)
void jpeg_dct_quant_kernel(const float* __restrict__ rgb,
                           const float* __restrict__ quant,
                           float* __restrict__ out) {
    __shared__ __align__(16) float lds[8 * 1024];      // 4 KB per wave
    __shared__ __align__(16) float qinv[192];          // 1/quant, shared by block
    const int lane = threadIdx.x & 31;
    const int wib  = threadIdx.x >> 5;
    const int wg   = blockIdx.x * 8 + wib;             // 32768 wave-tasks
    const int regX  = wg & 31;                         // 32 regions of 32 px
    const int regY  = (wg >> 5) & 63;                  // 64 regions of 16 px
    const int batch = wg >> 11;                        // 16 images

    // --- quantizer reciprocals, once per block (kills IEEE-div microcode) ---
    if (threadIdx.x < 192) qinv[threadIdx.x] = 1.0f / quant[threadIdx.x];
    __syncthreads();

    float* Yl = lds + wib * 1024;                      // 16 x 32 luma tile pair
    float* Ct = Yl + 512;                              // 16 x 16 chroma tile
    float* St = Ct + 256;                              // 16 x 16 S^T scratch

    const int px0 = regX << 5;
    const int py0 = regY << 4;
    const float* rb = rgb + (size_t)batch * 3u * 1024u * 1024u;
    const int x = px0 + lane;

    // ---- Phase 0: load RGB, convert, pool chroma 2x2 via shfl_xor ----
#pragma unroll
    for (int pr = 0; pr < 8; ++pr) {
        float cbs = 0.0f, crs = 0.0f;
#pragma unroll
        for (int s = 0; s < 2; ++s) {
            const int yy = py0 + 2 * pr + s;
            const size_t off = (size_t)yy * 1024u + (size_t)x;
            const float r = rb[off];
            const float g = rb[off + 1048576u];
            const float b = rb[off + 2097152u];
            const float Y  = fminf(fmaxf(0.299f * r + 0.587f * g + 0.114f * b, 0.0f), 1.0f) - JPEG_OFFSET;
            const float Cb = fminf(fmaxf(-0.168735892f * r - 0.331264108f * g + 0.5f * b + JPEG_OFFSET, 0.0f), 1.0f) - JPEG_OFFSET;
            const float Cr = fminf(fmaxf(0.5f * r - 0.418687589f * g - 0.081312411f * b + JPEG_OFFSET, 0.0f), 1.0f) - JPEG_OFFSET;
            Yl[(2 * pr + s) * 32 + lane] = Y;
            cbs += Cb; crs += Cr;
        }
        cbs += __shfl_xor(cbs, 1, 32);
        crs += __shfl_xor(crs, 1, 32);
        if ((lane & 1) == 0) {
            Ct[pr * 16 + (lane >> 1)]       = cbs * 0.25f;   // Cb pooled rows 0-7
            Ct[(8 + pr) * 16 + (lane >> 1)] = crs * 0.25f;   // Cr pooled rows 8-15
        }
    }
    __builtin_amdgcn_wave_barrier();

    const AFrag A = make_afrag(lane);
    const int tr = lane & 15, hi = lane >> 4;
    const int i = tr & 7;                              // in-block row

    // ---- Y tiles: block grid 128x128, out base 0 ----
    {
        const f4 qy0 = *(const f4*)(qinv + i * 8);
        const f4 qy1 = *(const f4*)(qinv + i * 8 + 4);
#pragma unroll
        for (int t = 0; t < 2; ++t) {
            v8f r = tile_dct(A, Yl + 16 * t, 32, St, lane);
            const int brow = (regY << 1) + (tr >> 3);
            const int bcol = (px0 >> 3) + 2 * t + hi;
            const size_t blk = ((size_t)batch * 128u + (size_t)brow) * 128u + (size_t)bcol;
            quant_store(r, qy0, qy1, out + blk * 64u + (size_t)i * 8u);
        }
    }

    // ---- Chroma tile: rows<8 -> Cb (ch1), rows>=8 -> Cr (ch2); grid 64x64 ----
    {
        v8f r = tile_dct(A, Ct, 16, St, lane);
        const int ch = 1 + (tr >> 3);
        const f4 qc0 = *(const f4*)(qinv + ch * 64 + i * 8);
        const f4 qc1 = *(const f4*)(qinv + ch * 64 + i * 8 + 4);
        const int bcol = (regX << 1) + hi;
        const size_t cblk = ((size_t)batch * 64u + (size_t)regY) * 64u + (size_t)bcol;
        const size_t base = (ch == 1) ? 16777216u : 20971520u;  // Y count / +Cb count
        quant_store(r, qc0, qc1, out + base + cblk * 64u + (size_t)i * 8u);
    }
}

extern "C" void kernel_launch(void* const* d_in, const int* in_sizes, int n_in,
                              void* d_out, int out_size, void* d_ws, size_t ws_size,
                              hipStream_t stream) {
    const float* rgb   = (const float*)d_in[0];   // (16,3,1024,1024) f32
    const float* quant = (const float*)d_in[1];   // (3,8,8) f32
    float* out = (float*)d_out;                   // Y | Cb | Cr blocks, f32
    (void)in_sizes; (void)n_in; (void)out_size; (void)d_ws; (void)ws_size;
    // 16 batches * 64 * 32 regions = 32768 wave-tasks, 8 waves (256 thr) per block
    jpeg_dct_quant_kernel<<<dim3(4096), dim3(256), 0, stream>>>(rgb, quant, out);
}